// FullyConnectedMLP_12584254177426
// MI455X (gfx1250) — compile-verified
//
#include <hip/hip_runtime.h>

typedef __attribute__((ext_vector_type(2))) float v2f;
typedef __attribute__((ext_vector_type(8))) float v8f;

#define NB     8192
#define NSTEPC 30
#define DD     100      // z_dim / assets
#define HH     110      // hidden
#define HP     112      // padded width (multiple of 16)
#define TT     29       // number of subnets
#define LDSS   114      // LDS row stride (even -> 8B-aligned b64 loads, conflict-free)
#define DT_C   (1.0f/30.0f)
#define R_C    0.05f
#define EPS_C  1e-3f

// reformatted weight strides (floats) per t:  nq * 2(hi) * HP(col) * 2(pair)
#define QSTR   (2 * HP * 2)        // 448 floats per k-step
#define S1R    (25 * QSTR)         // W1: K=100 -> 25 k-steps
#define S2R    (28 * QSTR)         // W2: K=110 -> 28 k-steps (padded)
#define S3R    (28 * QSTR)         // W3: K=110 -> 28 k-steps (padded)

// ---------------------------------------------------------------------------
// Weight reformat: R[t][q][hi][col][j] = W[t][4q+hi+2j][col], zero padded.
// Gives each lane its (b.x,b.y) WMMA B-fragment pair contiguous in memory.
// ---------------------------------------------------------------------------
__global__ void reformat_kernel(const float* __restrict__ W, float* __restrict__ R,
                                int Krows, int Ncols, int nq) {
  int idx = blockIdx.x * blockDim.x + threadIdx.x;
  int perT = nq * QSTR;
  int total = TT * perT;
  if (idx >= total) return;
  int t   = idx / perT;
  int rem = idx % perT;
  int j   = rem & 1;
  int col = (rem >> 1) % HP;
  int qh  = (rem >> 1) / HP;
  int hi  = qh & 1;
  int q   = qh >> 1;
  int krow = q * 4 + hi + 2 * j;
  float val = (krow < Krows && col < Ncols)
                  ? W[t * Krows * Ncols + krow * Ncols + col]
                  : 0.f;
  R[idx] = val;
}

// ---------------------------------------------------------------------------
// 32xN GEMM tile pair: two 16-row A tiles (LDS rows r and r+16) share each
// B fragment. REFMT: B pair via one b64 from the reformatted array with
// constant per-q offsets that fold into the instruction IOFFSET; else
// bounds-checked loads straight from row-major W.
// ---------------------------------------------------------------------------
template <bool REFMT>
__device__ __forceinline__ void gemm32(const float* sIn,
                                       const float* __restrict__ Wraw,
                                       const float* __restrict__ Wref,
                                       int Krows, int Ncols, int nq, int n0,
                                       int lane, v8f& acc0, v8f& acc1) {
  const int r   = lane & 15;
  const int hi  = (lane >> 4) & 1;
  const int khi = hi * 2;
  const int col = n0 + r;
  const float* ap0 = sIn + r * LDSS + khi;        // A tile 0 (rows 0..15)
  const float* ap1 = ap0 + 16 * LDSS;             // A tile 1 (rows 16..31)
  const float* bp  = Wref + (hi * HP + col) * 2;  // per-lane fragment base
  for (int q = 0; q < nq; ++q) {
    v2f a0 = *(const v2f*)(ap0 + q * 4);
    v2f a1 = *(const v2f*)(ap1 + q * 4);
    v2f b;
    if (REFMT) {
      b = *(const v2f*)(bp + q * QSTR);
    } else {
      int k = q * 4;
      int r0 = k + hi, r1 = r0 + 2;
      b.x = (r0 < Krows && col < Ncols) ? Wraw[r0 * Ncols + col] : 0.f;
      b.y = (r1 < Krows && col < Ncols) ? Wraw[r1 * Ncols + col] : 0.f;
    }
    acc0 = __builtin_amdgcn_wmma_f32_16x16x4_f32(false, a0, false, b,
                                                 (short)0, acc0, false, false);
    acc1 = __builtin_amdgcn_wmma_f32_16x16x4_f32(false, a1, false, b,
                                                 (short)0, acc1, false, false);
  }
}

// ---------------------------------------------------------------------------
// Init: out[b] = y_init*c^30 + c^29 * sum_d sig*X[b,0,d]*z_init[d]*DW[b,0,d]
// ---------------------------------------------------------------------------
__global__ void bsde_init_kernel(const float* __restrict__ X,
                                 const float* __restrict__ DWs,
                                 const float* __restrict__ sig,
                                 const float* __restrict__ yinit,
                                 const float* __restrict__ zinit,
                                 float* __restrict__ out) {
  int b = blockIdx.x * blockDim.x + threadIdx.x;
  if (b >= NB) return;
  const float c = 1.0f + R_C * DT_C;
  int base = b * NSTEPC * DD;
  float s = 0.f;
  for (int d = 0; d < DD; ++d)
    s += sig[d] * X[base + d] * zinit[d] * DWs[base + d];
  out[b] = yinit[0] * __powf(c, 30.f) + __powf(c, 29.f) * s;
}

// ---------------------------------------------------------------------------
// Block = 2 waves, 32 batch rows, one subnet t. Wave w handles N-tiles
// n = w, w+2, w+4, w+6 of each layer; activations ping-pong through LDS.
// Then out[row] += c^(28-t) * sum_d sig*X[:,jX,d]*Z[:,d]*DW[:,jW,d].
// ---------------------------------------------------------------------------
template <bool REFMT>
__global__ __launch_bounds__(64) void bsde_step_kernel(
    const float* __restrict__ X,  const float* __restrict__ DWs,
    const float* __restrict__ sig,
    const float* __restrict__ W1, const float* __restrict__ W2,
    const float* __restrict__ W3,
    const float* __restrict__ R1, const float* __restrict__ R2,
    const float* __restrict__ R3,
    const float* __restrict__ g0, const float* __restrict__ b0,
    const float* __restrict__ m0v, const float* __restrict__ v0,
    const float* __restrict__ g1, const float* __restrict__ b1,
    const float* __restrict__ m1, const float* __restrict__ v1,
    const float* __restrict__ g2, const float* __restrict__ b2,
    const float* __restrict__ m2, const float* __restrict__ v2,
    const float* __restrict__ g3, const float* __restrict__ b3,
    const float* __restrict__ m3, const float* __restrict__ v3,
    float* __restrict__ out) {
  __shared__ float sA[32 * LDSS];
  __shared__ float sB[32 * LDSS];
  __shared__ float sAcc[32];

  const int tid   = threadIdx.x;
  const int wave  = tid >> 5;
  const int lane  = tid & 31;
  const int mrow0 = blockIdx.x * 32;
  const int t     = blockIdx.y;          // subnet index 0..28

  const float* w1 = W1 + t * (DD * HH);
  const float* w2 = W2 + t * (HH * HH);
  const float* w3 = W3 + t * (HH * DD);
  const float* r1 = R1 + t * S1R;
  const float* r2 = R2 + t * S2R;
  const float* r3 = R3 + t * S3R;

  if (tid < 32) sAcc[tid] = 0.f;

  // ---- stage BN0(X[:, t+1, :]) into sA: column-per-thread, coalesced ----
  const int stepin = t + 1;
  for (int d = tid; d < HP; d += 64) {
    float sc = 0.f, sh = 0.f;
    if (d < DD) {
      sc = g0[t * DD + d] * rsqrtf(v0[t * DD + d] + EPS_C);
      sh = b0[t * DD + d] - m0v[t * DD + d] * sc;
    }
    const float* xp = X + (mrow0 * NSTEPC + stepin) * DD + d;
    for (int r = 0; r < 32; ++r) {
      float val = 0.f;
      if (d < DD) val = xp[r * (NSTEPC * DD)] * sc + sh;
      sA[r * LDSS + d] = val;
    }
  }
  __syncthreads();

  const int hi   = (lane >> 4) & 1;
  const int colL = lane & 15;

  // ---- layer 1: sB = relu(BN1(sA @ w1)), K=100(25q), N=110 ----
#pragma unroll 1
  for (int n = wave; n < 7; n += 2) {
    int n0 = n * 16;
    v8f acc0 = {0.f,0.f,0.f,0.f,0.f,0.f,0.f,0.f};
    v8f acc1 = acc0;
    gemm32<REFMT>(sA, w1, r1, DD, HH, 25, n0, lane, acc0, acc1);
    int col = n0 + colL;
    float sc = 0.f, sh = 0.f;
    if (col < HH) {
      sc = g1[t * HH + col] * rsqrtf(v1[t * HH + col] + EPS_C);
      sh = b1[t * HH + col] - m1[t * HH + col] * sc;
    }
    union { v8f v; float f[8]; } u0, u1; u0.v = acc0; u1.v = acc1;
    for (int jj = 0; jj < 8; ++jj) {
      int row = jj + 8 * hi;
      float v0f = u0.f[jj] * sc + sh; v0f = v0f > 0.f ? v0f : 0.f;
      float v1f = u1.f[jj] * sc + sh; v1f = v1f > 0.f ? v1f : 0.f;
      sB[row * LDSS + col]        = (col < HH) ? v0f : 0.f;
      sB[(row + 16) * LDSS + col] = (col < HH) ? v1f : 0.f;
    }
  }
  __syncthreads();

  // ---- layer 2: sA = relu(BN2(sB @ w2)), K=110(28q pad), N=110 ----
#pragma unroll 1
  for (int n = wave; n < 7; n += 2) {
    int n0 = n * 16;
    v8f acc0 = {0.f,0.f,0.f,0.f,0.f,0.f,0.f,0.f};
    v8f acc1 = acc0;
    gemm32<REFMT>(sB, w2, r2, HH, HH, 28, n0, lane, acc0, acc1);
    int col = n0 + colL;
    float sc = 0.f, sh = 0.f;
    if (col < HH) {
      sc = g2[t * HH + col] * rsqrtf(v2[t * HH + col] + EPS_C);
      sh = b2[t * HH + col] - m2[t * HH + col] * sc;
    }
    union { v8f v; float f[8]; } u0, u1; u0.v = acc0; u1.v = acc1;
    for (int jj = 0; jj < 8; ++jj) {
      int row = jj + 8 * hi;
      float v0f = u0.f[jj] * sc + sh; v0f = v0f > 0.f ? v0f : 0.f;
      float v1f = u1.f[jj] * sc + sh; v1f = v1f > 0.f ? v1f : 0.f;
      sA[row * LDSS + col]        = (col < HH) ? v0f : 0.f;
      sA[(row + 16) * LDSS + col] = (col < HH) ? v1f : 0.f;
    }
  }
  __syncthreads();

  // ---- layer 3: sB = BN3(sA @ w3), K=110(28q pad), N=100 ----
#pragma unroll 1
  for (int n = wave; n < 7; n += 2) {
    int n0 = n * 16;
    v8f acc0 = {0.f,0.f,0.f,0.f,0.f,0.f,0.f,0.f};
    v8f acc1 = acc0;
    gemm32<REFMT>(sA, w3, r3, HH, DD, 28, n0, lane, acc0, acc1);
    int col = n0 + colL;
    float sc = 0.f, sh = 0.f;
    if (col < DD) {
      sc = g3[t * DD + col] * rsqrtf(v3[t * DD + col] + EPS_C);
      sh = b3[t * DD + col] - m3[t * DD + col] * sc;
    }
    union { v8f v; float f[8]; } u0, u1; u0.v = acc0; u1.v = acc1;
    for (int jj = 0; jj < 8; ++jj) {
      int row = jj + 8 * hi;
      sB[row * LDSS + col]        = (col < DD) ? (u0.f[jj] * sc + sh) : 0.f;
      sB[(row + 16) * LDSS + col] = (col < DD) ? (u1.f[jj] * sc + sh) : 0.f;
    }
  }
  __syncthreads();

  // ---- epilogue: 2 threads per row, 50 contiguous cols each, 1 LDS atomic ----
  const int jX = (t == TT - 1) ? (NSTEPC - 2) : (t + 1);
  const int jW = (t == TT - 1) ? (NSTEPC - 1) : (t + 1);
  {
    const int er  = tid >> 1;          // row 0..31
    const int off = (tid & 1) * 50;    // column half
    const int xbase = (mrow0 + er) * (NSTEPC * DD);
    const float* xp = X   + xbase + jX * DD + off;
    const float* wp = DWs + xbase + jW * DD + off;
    const float* zp = sB  + er * LDSS + off;
    const float* sp = sig + off;
    float partial = 0.f;
    for (int i = 0; i < 50; ++i)
      partial += sp[i] * xp[i] * wp[i] * zp[i];
    atomicAdd(&sAcc[er], partial);
  }
  __syncthreads();
  if (tid < 32) {
    const float c = 1.0f + R_C * DT_C;
    const float factor = __powf(c, (float)(TT - 1 - t));
    atomicAdd(&out[mrow0 + tid], factor * sAcc[tid]);
  }
}

// ---------------------------------------------------------------------------
extern "C" void kernel_launch(void* const* d_in, const int* in_sizes, int n_in,
                              void* d_out, int out_size, void* d_ws, size_t ws_size,
                              hipStream_t stream) {
  (void)in_sizes; (void)n_in; (void)out_size;
  const float* X     = (const float*)d_in[0];
  const float* DWs   = (const float*)d_in[1];
  const float* sig   = (const float*)d_in[2];
  const float* yinit = (const float*)d_in[3];
  const float* zinit = (const float*)d_in[4];
  const float* W1    = (const float*)d_in[5];
  const float* W2    = (const float*)d_in[6];
  const float* W3    = (const float*)d_in[7];
  const float* g0 = (const float*)d_in[8],  *b0 = (const float*)d_in[9];
  const float* m0 = (const float*)d_in[10], *v0 = (const float*)d_in[11];
  const float* g1 = (const float*)d_in[12], *b1 = (const float*)d_in[13];
  const float* m1 = (const float*)d_in[14], *v1 = (const float*)d_in[15];
  const float* g2 = (const float*)d_in[16], *b2 = (const float*)d_in[17];
  const float* m2 = (const float*)d_in[18], *v2 = (const float*)d_in[19];
  const float* g3 = (const float*)d_in[20], *b3 = (const float*)d_in[21];
  const float* m3 = (const float*)d_in[22], *v3 = (const float*)d_in[23];
  float* out = (float*)d_out;

  const size_t need = (size_t)TT * (S1R + S2R + S3R) * sizeof(float);
  const bool refmt = (ws_size >= need);
  float* R1 = (float*)d_ws;
  float* R2 = R1 + TT * S1R;
  float* R3 = R2 + TT * S2R;

  if (refmt) {
    int n1 = TT * S1R, n2 = TT * S2R, n3 = TT * S3R;
    reformat_kernel<<<(n1 + 255) / 256, 256, 0, stream>>>(W1, R1, DD, HH, 25);
    reformat_kernel<<<(n2 + 255) / 256, 256, 0, stream>>>(W2, R2, HH, HH, 28);
    reformat_kernel<<<(n3 + 255) / 256, 256, 0, stream>>>(W3, R3, HH, DD, 28);
  }

  bsde_init_kernel<<<NB / 256, 256, 0, stream>>>(X, DWs, sig, yinit, zinit, out);

  dim3 grid(NB / 32, TT);
  if (refmt) {
    bsde_step_kernel<true><<<grid, 64, 0, stream>>>(
        X, DWs, sig, W1, W2, W3, R1, R2, R3,
        g0, b0, m0, v0, g1, b1, m1, v1,
        g2, b2, m2, v2, g3, b3, m3, v3, out);
  } else {
    bsde_step_kernel<false><<<grid, 64, 0, stream>>>(
        X, DWs, sig, W1, W2, W3, R1, R2, R3,
        g0, b0, m0, v0, g1, b1, m1, v1,
        g2, b2, m2, v2, g3, b3, m3, v3, out);
  }
}